// QuanvolutionClassifier_65481071400706
// MI455X (gfx1250) — compile-verified
//
#include <hip/hip_runtime.h>

typedef float v2f __attribute__((ext_vector_type(2)));
typedef float v8f __attribute__((ext_vector_type(8)));

#define N_PATCH 196

// ---------------------------------------------------------------------------
// Kernel 0: build the fixed ansatz operator A (16x16) from params and store it
// in V_WMMA_F32_16X16X4_F32 A-operand layout chunks:
//   chunk c, lane L: M = L&15, K = 4c + (L>>4)*2 .. +1   -> float2 per lane.
// ---------------------------------------------------------------------------
__global__ void qc_prep(const float* __restrict__ params, float2* __restrict__ wsA) {
  __shared__ float Amat[16][16];
  const int L = threadIdx.x;
  if (L < 16) {
    float col[16];
#pragma unroll
    for (int s = 0; s < 16; ++s) col[s] = (s == L) ? 1.f : 0.f;
#pragma unroll
    for (int l = 0; l < 4; ++l) {
#pragma unroll
      for (int w = 0; w < 4; ++w) {
        float th = params[l * 4 + w] * 0.5f;
        float cc = cosf(th), ss = sinf(th);
        const int mask = 8 >> w;   // wire w <-> bit (3-w) of state index
#pragma unroll
        for (int a = 0; a < 16; ++a) {
          if (!(a & mask)) {
            int bidx = a | mask;
            float x0 = col[a], x1 = col[bidx];
            col[a]    = cc * x0 - ss * x1;
            col[bidx] = ss * x0 + cc * x1;
          }
        }
      }
      // CX(0,1): rows with wire0 bit (mask 8): s <-> s^4
#pragma unroll
      for (int t = 8; t < 12; ++t) { float tmp = col[t]; col[t] = col[t + 4]; col[t + 4] = tmp; }
      // CX(2,3): rows with wire2 bit (mask 2): s <-> s^1
#pragma unroll
      for (int t = 0; t < 16; t += 4) { float tmp = col[t + 2]; col[t + 2] = col[t + 3]; col[t + 3] = tmp; }
    }
#pragma unroll
    for (int s = 0; s < 16; ++s) Amat[s][L] = col[s];
  }
  __syncthreads();
  const int m   = L & 15;
  const int off = (L >> 4) * 2;
#pragma unroll
  for (int c = 0; c < 4; ++c) {
    int kk = 4 * c + off;
    float2 v; v.x = Amat[m][kk]; v.y = Amat[m][kk + 1];
    wsA[c * 32 + L] = v;
  }
}

// ---------------------------------------------------------------------------
// Kernel 1: one wave per image. 13 rounds of 16 patches; per round the wave
// builds V (16 states x 16 patches) in B-operand layout and computes
// U = A*V with 4 chained f32 WMMAs, then probs/expectations + feat store.
// ---------------------------------------------------------------------------
__global__ void qc_quanv(const float* __restrict__ x, const float2* __restrict__ wsA,
                         float* __restrict__ feat, int bsz) {
  const int L   = threadIdx.x & 31;
  const int img = (int)((blockIdx.x * blockDim.x + threadIdx.x) >> 5);
  if (img >= bsz) return;                       // uniform per wave

  v2f A[4];
#pragma unroll
  for (int c = 0; c < 4; ++c) { float2 t = wsA[c * 32 + L]; A[c].x = t.x; A[c].y = t.y; }

  const float* xb = x + (size_t)img * 784;
  const int n = L & 15;                         // patch within the round (both halves)
  const int h = L >> 4;                         // 0: supplies K=0,1 / states 0-7; 1: K=2,3 / states 8-15

  for (int rr = 0; rr < 13; ++rr) {
    int  p     = rr * 16 + n;
    bool valid = (p < N_PATCH);
    int  pcl   = valid ? p : (N_PATCH - 1);     // clamp to stay in-bounds
    int  pr    = pcl / 14;
    int  pc    = pcl - pr * 14;
    int  base  = pr * 56 + pc * 2;              // (2*pr)*28 + 2*pc
    float2 top = *(const float2*)(xb + base);        // pixels [r,c],[r,c+1]   -> wires 0,1
    float2 bot = *(const float2*)(xb + base + 28);   // pixels [r+1,c],[r+1,c+1]-> wires 2,3

    float c0, s0, c1, s1, c2, s2, c3, s3;
    __sincosf(top.x * 0.5f, &s0, &c0);
    __sincosf(top.y * 0.5f, &s1, &c1);
    __sincosf(bot.x * 0.5f, &s2, &c2);
    __sincosf(bot.y * 0.5f, &s3, &c3);

    // v[s] = t0[s>>3] * t1[(s>>2)&1] * t2[(s>>1)&1] * t3[s&1]
    float q0 = c0 * c1, q1 = c0 * s1, q2 = s0 * c1, q3 = s0 * s1;  // wires 0,1
    float r0 = c2 * c3, r1 = c2 * s3, r2 = s2 * c3, r3 = s2 * s3;  // wires 2,3
    float qv[4] = {q0, q1, q2, q3};
    float rv[4] = {r0, r1, r2, r3};
    float v[16];
#pragma unroll
    for (int s = 0; s < 16; ++s) v[s] = qv[s >> 2] * rv[s & 3];

    v8f acc = {0.f, 0.f, 0.f, 0.f, 0.f, 0.f, 0.f, 0.f};
#pragma unroll
    for (int c = 0; c < 4; ++c) {
      v2f bop;
      bop.x = h ? v[4 * c + 2] : v[4 * c + 0];
      bop.y = h ? v[4 * c + 3] : v[4 * c + 1];
      acc = __builtin_amdgcn_wmma_f32_16x16x4_f32(false, A[c], false, bop,
                                                  (short)0, acc, false, false);
    }

    // C layout: VGPR j holds state (j + 8h) for patch n. probs + signed sums.
    float pb[8];
#pragma unroll
    for (int j = 0; j < 8; ++j) pb[j] = acc[j] * acc[j];
    float t01 = pb[0] + pb[1], t23 = pb[2] + pb[3], t45 = pb[4] + pb[5], t67 = pb[6] + pb[7];
    float d01 = pb[0] - pb[1], d23 = pb[2] - pb[3], d45 = pb[4] - pb[5], d67 = pb[6] - pb[7];
    float e3 = (d01 + d23) + (d45 + d67);       // wire3: sign by bit0
    float e2 = (t01 - t23) + (t45 - t67);       // wire2: sign by bit1
    float s0123 = t01 + t23, s4567 = t45 + t67;
    float e1 = s0123 - s4567;                   // wire1: sign by bit2
    float tot = s0123 + s4567;
    float e0 = h ? -tot : tot;                  // wire0: sign by bit3 (= half)
    e0 += __shfl_xor(e0, 16, 32);
    e1 += __shfl_xor(e1, 16, 32);
    e2 += __shfl_xor(e2, 16, 32);
    e3 += __shfl_xor(e3, 16, 32);

    if (h == 0 && valid) {
      float4 o; o.x = e0; o.y = e1; o.z = e2; o.w = e3;
      *(float4*)(feat + (size_t)img * 784 + p * 4) = o;
    }
  }
}

// ---------------------------------------------------------------------------
// Kernel 2: one wave per 16 images. logits(16x16) = feat(16x784) x W^T(784x16pad)
// via 196 f32 WMMAs, + bias, masked log_softmax across 16-lane class groups.
// ---------------------------------------------------------------------------
__global__ void qc_head(const float* __restrict__ feat, const float* __restrict__ W,
                        const float* __restrict__ b, float* __restrict__ out, int bsz) {
  const int L    = threadIdx.x & 31;
  const int wave = (int)((blockIdx.x * blockDim.x + threadIdx.x) >> 5);
  const int imgbase = wave * 16;
  if (imgbase >= bsz) return;                   // uniform per wave

  const int m   = L & 15;                       // A: image row; B/C: class column
  const int h   = L >> 4;
  const int off = h * 2;
  int mi = imgbase + m; if (mi > bsz - 1) mi = bsz - 1;
  const float* fm  = feat + (size_t)mi * 784;
  const bool  isCls = (m < 10);
  const float* wr  = W + (size_t)(isCls ? m : 0) * 784;

  v8f acc = {0.f, 0.f, 0.f, 0.f, 0.f, 0.f, 0.f, 0.f};
#pragma unroll 4
  for (int c = 0; c < 196; ++c) {
    const float2 af = *(const float2*)(fm + 4 * c + off);
    const float2 bf = *(const float2*)(wr + 4 * c + off);
    v2f aop; aop.x = af.x; aop.y = af.y;
    v2f bop; bop.x = isCls ? bf.x : 0.f; bop.y = isCls ? bf.y : 0.f;
    acc = __builtin_amdgcn_wmma_f32_16x16x4_f32(false, aop, false, bop,
                                                (short)0, acc, false, false);
  }

  const float bb = isCls ? b[m] : 0.f;
#pragma unroll
  for (int j = 0; j < 8; ++j) {
    int   imgj = imgbase + j + 8 * h;           // C: VGPR j, lanes>=16 hold M=j+8
    float lg   = acc[j] + bb;
    float vmax = isCls ? lg : -3.0e38f;
#pragma unroll
    for (int d = 1; d < 16; d <<= 1) vmax = fmaxf(vmax, __shfl_xor(vmax, d, 16));
    float ex  = isCls ? __expf(lg - vmax) : 0.f;
    float sum = ex;
#pragma unroll
    for (int d = 1; d < 16; d <<= 1) sum += __shfl_xor(sum, d, 16);
    float lp = lg - vmax - __logf(sum);
    if (isCls && imgj < bsz) out[(size_t)imgj * 10 + m] = lp;
  }
}

// ---------------------------------------------------------------------------
extern "C" void kernel_launch(void* const* d_in, const int* in_sizes, int n_in,
                              void* d_out, int out_size, void* d_ws, size_t ws_size,
                              hipStream_t stream) {
  const float* x      = (const float*)d_in[0];   // (bsz, 784)
  const float* params = (const float*)d_in[1];   // (4, 4)
  const float* W      = (const float*)d_in[2];   // (10, 784)
  const float* b      = (const float*)d_in[3];   // (10,)
  float*       out    = (float*)d_out;           // (bsz, 10)

  const int bsz = in_sizes[0] / 784;

  float2* wsA  = (float2*)d_ws;                          // 1 KB: A-operand chunks
  float*  feat = (float*)((char*)d_ws + 1024);           // bsz*784 floats

  qc_prep<<<1, 32, 0, stream>>>(params, wsA);

  int waves1  = bsz;                                     // one wave per image
  int blocks1 = (waves1 * 32 + 255) / 256;
  qc_quanv<<<blocks1, 256, 0, stream>>>(x, wsA, feat, bsz);

  int waves2  = (bsz + 15) / 16;                         // one wave per 16 images
  int blocks2 = (waves2 * 32 + 255) / 256;
  qc_head<<<blocks2, 256, 0, stream>>>(feat, W, b, out, bsz);
}